// GNN_46110768890112
// MI455X (gfx1250) — compile-verified
//
#include <hip/hip_runtime.h>

typedef __attribute__((ext_vector_type(2))) float v2f;
typedef __attribute__((ext_vector_type(8))) float v8f;

#define N_NODES  100000
#define N_EDGES  1600000
#define N_GRAPHS 256
#define F_IN     38
#define HID      64

static __host__ __device__ inline size_t align256(size_t b) { return (b + 255) & ~(size_t)255; }

// ---------------------------------------------------------------------------
// Edge scatter, layer 1: agg1[dst] += x[src]  (38 features, fp32 atomics; all
// buffers L2-resident on MI455X's 192MB L2). One thread per edge, float2 loads
// (row byte stride 152 is 8B aligned).
// ---------------------------------------------------------------------------
__global__ __launch_bounds__(256) void scatter_agg38(const float* __restrict__ x,
                                                     const int* __restrict__ ei,
                                                     float* __restrict__ agg) {
    int e = blockIdx.x * blockDim.x + threadIdx.x;
    if (e >= N_EDGES) return;
    int src = ei[e];
    int dst = ei[N_EDGES + e];
    const float* xr = x + (size_t)src * F_IN;
    float* ar = agg + (size_t)dst * F_IN;
#pragma unroll
    for (int k = 0; k < F_IN; k += 2) {
        v2f v = *(const v2f*)(xr + k);
        atomicAdd(ar + k,     v.x);
        atomicAdd(ar + k + 1, v.y);
    }
}

// ---------------------------------------------------------------------------
// Edge scatter, layer 2: agg2[dst] += h[src]  (64 features, float4 loads).
// ---------------------------------------------------------------------------
__global__ __launch_bounds__(256) void scatter_agg64(const float* __restrict__ h,
                                                     const int* __restrict__ ei,
                                                     float* __restrict__ agg) {
    int e = blockIdx.x * blockDim.x + threadIdx.x;
    if (e >= N_EDGES) return;
    int src = ei[e];
    int dst = ei[N_EDGES + e];
    const float4* hr = (const float4*)(h + (size_t)src * HID);
    float* ar = agg + (size_t)dst * HID;
#pragma unroll
    for (int k = 0; k < HID / 4; ++k) {
        float4 v = hr[k];
        atomicAdd(ar + 4 * k + 0, v.x);
        atomicAdd(ar + 4 * k + 1, v.y);
        atomicAdd(ar + 4 * k + 2, v.z);
        atomicAdd(ar + 4 * k + 3, v.w);
    }
}

// ---------------------------------------------------------------------------
// Layer 1 GEMM: h = relu([agg1 | x] @ [w1_rel ; w1_root] + b1), K = 76.
// Per block: 16 rows x 64 cols, 4 waves, one v_wmma_f32_16x16x4_f32 tile each.
// N_NODES % 16 == 0 -> no guards -> EXEC all ones at every WMMA (ISA req).
// ---------------------------------------------------------------------------
__global__ __launch_bounds__(128) void gemm1_relu(const float* __restrict__ agg,
                                                  const float* __restrict__ x,
                                                  const float* __restrict__ w_rel,
                                                  const float* __restrict__ b_rel,
                                                  const float* __restrict__ w_root,
                                                  float* __restrict__ h) {
    __shared__ float As[16][80];                     // 76 used, padded
    const int tid  = threadIdx.x;
    const int lane = tid & 31;
    const int wave = tid >> 5;                       // 0..3 -> 16-col slice
    const int r0   = blockIdx.x * 16;

    for (int i = tid; i < 16 * 76; i += 128) {       // stage A tile in LDS
        int r = i / 76, k = i % 76;
        As[r][k] = (k < F_IN) ? agg[(size_t)(r0 + r) * F_IN + k]
                              : x[(size_t)(r0 + r) * F_IN + (k - F_IN)];
    }
    __syncthreads();

    const int mrow  = lane & 15;
    const int khalf = lane >> 4;                     // 0: K=k0,k0+1  1: K=k0+2,k0+3
    const int col   = wave * 16 + mrow;

    v8f acc = {};
#pragma unroll
    for (int kk = 0; kk < 19; ++kk) {                // K = 76 = 19 * 4
        const int k0 = kk * 4 + 2 * khalf;
        v2f a;
        a.x = As[mrow][k0];
        a.y = As[mrow][k0 + 1];
        v2f b;
        b.x = (k0     < F_IN) ? w_rel[(k0    ) * HID + col] : w_root[(k0     - F_IN) * HID + col];
        b.y = (k0 + 1 < F_IN) ? w_rel[(k0 + 1) * HID + col] : w_root[(k0 + 1 - F_IN) * HID + col];
        acc = __builtin_amdgcn_wmma_f32_16x16x4_f32(false, a, false, b, (short)0, acc, false, false);
    }

    const float bias = b_rel[col];
#pragma unroll
    for (int r = 0; r < 8; ++r) {                    // C layout: row r / r+8
        int row = r0 + r + khalf * 8;
        float v = acc[r] + bias;
        h[(size_t)row * HID + col] = v > 0.f ? v : 0.f;
    }
}

// ---------------------------------------------------------------------------
// Layer 2 GEMM fused with mean-pool scatter:
//   h2 = [agg2 | h] @ [w2_rel ; w2_root] + b2  (K = 128), then
//   sums[batch[row]] += h2[row]  (atomic, 64KB array -> L2/WGP$ hot).
// ---------------------------------------------------------------------------
__global__ __launch_bounds__(128) void gemm2_pool(const float* __restrict__ agg,
                                                  const float* __restrict__ h,
                                                  const float* __restrict__ w_rel,
                                                  const float* __restrict__ b_rel,
                                                  const float* __restrict__ w_root,
                                                  const int* __restrict__ batch,
                                                  float* __restrict__ sums) {
    __shared__ float As[16][132];                    // 128 used, padded
    const int tid  = threadIdx.x;
    const int lane = tid & 31;
    const int wave = tid >> 5;
    const int r0   = blockIdx.x * 16;

    for (int i = tid; i < 16 * 128; i += 128) {
        int r = i >> 7, k = i & 127;
        As[r][k] = (k < HID) ? agg[(size_t)(r0 + r) * HID + k]
                             : h[(size_t)(r0 + r) * HID + (k - HID)];
    }
    __syncthreads();

    const int mrow  = lane & 15;
    const int khalf = lane >> 4;
    const int col   = wave * 16 + mrow;

    v8f acc = {};
#pragma unroll
    for (int kk = 0; kk < 32; ++kk) {                // K = 128 = 32 * 4
        const int k0 = kk * 4 + 2 * khalf;
        v2f a;
        a.x = As[mrow][k0];
        a.y = As[mrow][k0 + 1];
        v2f b;
        b.x = (k0     < HID) ? w_rel[(k0    ) * HID + col] : w_root[(k0     - HID) * HID + col];
        b.y = (k0 + 1 < HID) ? w_rel[(k0 + 1) * HID + col] : w_root[(k0 + 1 - HID) * HID + col];
        acc = __builtin_amdgcn_wmma_f32_16x16x4_f32(false, a, false, b, (short)0, acc, false, false);
    }

    const float bias = b_rel[col];
#pragma unroll
    for (int r = 0; r < 8; ++r) {
        int row = r0 + r + khalf * 8;
        float v = acc[r] + bias;
        atomicAdd(&sums[(size_t)batch[row] * HID + col], v);
    }
}

__global__ __launch_bounds__(256) void count_nodes(const int* __restrict__ batch,
                                                   float* __restrict__ counts) {
    int i = blockIdx.x * blockDim.x + threadIdx.x;
    if (i < N_NODES) atomicAdd(&counts[batch[i]], 1.0f);
}

__global__ __launch_bounds__(256) void finalize(const float* __restrict__ sums,
                                                const float* __restrict__ counts,
                                                float* __restrict__ out) {
    int i = blockIdx.x * blockDim.x + threadIdx.x;
    if (i < N_GRAPHS * HID) {
        float c = counts[i >> 6];
        float denom = c > 1.f ? c : 1.f;
        float v = sums[i] / denom;
        out[i] = v > 0.f ? v : 0.f;
    }
}

extern "C" void kernel_launch(void* const* d_in, const int* in_sizes, int n_in,
                              void* d_out, int out_size, void* d_ws, size_t ws_size,
                              hipStream_t stream) {
    (void)in_sizes; (void)n_in; (void)out_size; (void)ws_size;
    const float* x       = (const float*)d_in[0];
    const int*   ei      = (const int*)d_in[1];
    const int*   batch   = (const int*)d_in[2];
    const float* w1_rel  = (const float*)d_in[3];
    const float* b1_rel  = (const float*)d_in[4];
    const float* w1_root = (const float*)d_in[5];
    const float* w2_rel  = (const float*)d_in[6];
    const float* b2_rel  = (const float*)d_in[7];
    const float* w2_root = (const float*)d_in[8];
    float* out = (float*)d_out;

    char* ws = (char*)d_ws;
    size_t o_agg1 = 0;
    size_t o_h    = o_agg1 + align256((size_t)N_NODES * F_IN * 4);
    size_t o_agg2 = o_h    + align256((size_t)N_NODES * HID * 4);
    size_t o_sums = o_agg2 + align256((size_t)N_NODES * HID * 4);
    size_t o_cnts = o_sums + align256((size_t)N_GRAPHS * HID * 4);
    float* agg1  = (float*)(ws + o_agg1);
    float* h     = (float*)(ws + o_h);
    float* agg2  = (float*)(ws + o_agg2);
    float* sums  = (float*)(ws + o_sums);
    float* cnts  = (float*)(ws + o_cnts);

    // Accumulators must be zero every call (atomics). Memset nodes are
    // graph-capturable; h is fully overwritten so no zeroing needed.
    hipMemsetAsync(agg1, 0, (size_t)N_NODES * F_IN * 4, stream);
    hipMemsetAsync(agg2, 0, (size_t)N_NODES * HID * 4, stream);
    hipMemsetAsync(sums, 0, (size_t)N_GRAPHS * HID * 4, stream);
    hipMemsetAsync(cnts, 0, (size_t)N_GRAPHS * 4, stream);

    scatter_agg38<<<(N_EDGES + 255) / 256, 256, 0, stream>>>(x, ei, agg1);
    gemm1_relu<<<N_NODES / 16, 128, 0, stream>>>(agg1, x, w1_rel, b1_rel, w1_root, h);
    scatter_agg64<<<(N_EDGES + 255) / 256, 256, 0, stream>>>(h, ei, agg2);
    count_nodes<<<(N_NODES + 255) / 256, 256, 0, stream>>>(batch, cnts);
    gemm2_pool<<<N_NODES / 16, 128, 0, stream>>>(agg2, h, w2_rel, b2_rel, w2_root, batch, sums);
    finalize<<<(N_GRAPHS * HID + 255) / 256, 256, 0, stream>>>(sums, cnts, out);
}